// GALinearAttentionModel_63745904607545
// MI455X (gfx1250) — compile-verified
//
#include <hip/hip_runtime.h>
#include <hip/hip_bf16.h>

typedef float v2f __attribute__((ext_vector_type(2)));
typedef float v8f __attribute__((ext_vector_type(8)));

namespace {
constexpr int B_ = 16;
constexpr int L_ = 4096;
constexpr int H_ = 128;
constexpr int DM_ = 11;
constexpr int LOOK = 8;
constexpr int T_ = L_ - LOOK;                       // 4088
constexpr int TILE_T = 24;
constexpr int ROWS = TILE_T + LOOK;                 // 32
constexpr int NTILES = (T_ + TILE_T - 1) / TILE_T;  // 171
constexpr int NPRED = B_ * T_;                      // 65408
constexpr float EPSF = 1e-6f;

// workspace layout (float offsets)
constexpr int WSO_WV   = 0;                  // wv_head[16]
constexpr int WSO_HM   = 16;                 // h_mask[3*128]
constexpr int WSO_PV   = 400;                // per-block |vec| partials [256]
constexpr int WSO_PB   = 656;                // per-block |biv| partials [256]
constexpr int WSO_QP   = 912;                // qonly partials [16*3]
constexpr int WSO_PRED = 960;                // base predictions [NPRED]
constexpr int WSO_DIFF = WSO_PRED + NPRED;   // 3 * NPRED ablation |diff|
}

__device__ __forceinline__ float phi_f(float v) {
  return (v >= 0.f ? v : 0.01f * v) + 1.f;
}

// Build one embedded row M (Dm=11, padded to 16 with zeros).
// zmask bit0: zero scalar ch, bit1: zero vector chs, bit2: zero bivector chs.
__device__ __forceinline__ void embed_row(const float xr[4], float sb,
                                          const float* vs, const float* vb,
                                          const float* bw, int zmask,
                                          float out[16]) {
  const int pi[6] = {0, 0, 0, 1, 1, 2};
  const int pj[6] = {1, 2, 3, 2, 3, 3};
  out[0] = (zmask & 1) ? 0.f : (1.f + sb);
#pragma unroll
  for (int c = 0; c < 4; ++c)
    out[1 + c] = (zmask & 2) ? 0.f : (xr[c] * vs[c] + vb[c]);
#pragma unroll
  for (int k = 0; k < 6; ++k)
    out[5 + k] = (zmask & 4) ? 0.f : (bw[k] * (xr[pi[k]] - xr[pj[k]]));
#pragma unroll
  for (int c = DM_; c < 16; ++c) out[c] = 0.f;
}

// ---------------------------------------------------------------------------
// prep: wv_head[d] = sum_h WV[h,d]*head_W[h];  h_mask for 3 channel ranges
// ---------------------------------------------------------------------------
__global__ void __launch_bounds__(128)
ga_prep_kernel(const float* __restrict__ WQ, const float* __restrict__ WV,
               const float* __restrict__ hW, float* __restrict__ ws) {
  __shared__ float red[128];
  const int h = threadIdx.x;
  if (h < 16) {
    float s = 0.f;
    if (h < DM_)
      for (int j = 0; j < H_; ++j) s += WV[j * DM_ + h] * hW[j];
    ws[WSO_WV + h] = s;
  }
  const int s0[3] = {0, 1, 5};
  const int s1[3] = {1, 5, DM_};
  for (int r = 0; r < 3; ++r) {
    float imp = 0.f;
    for (int d = s0[r]; d < s1[r]; ++d) imp += fabsf(WQ[h * DM_ + d]);
    red[h] = imp;
    __syncthreads();
    for (int off = 64; off > 0; off >>= 1) {
      if (h < off) red[h] = fmaxf(red[h], red[h + off]);
      __syncthreads();
    }
    const float mx = red[0];
    __syncthreads();
    ws[WSO_HM + r * H_ + h] = imp / (mx + 1e-12f);
  }
}

// ---------------------------------------------------------------------------
// magnitudes of vector / bivector channels (per-block deterministic partials)
// ---------------------------------------------------------------------------
__global__ void __launch_bounds__(256)
ga_mag_kernel(const float* __restrict__ x, const float* __restrict__ vs,
              const float* __restrict__ vb, const float* __restrict__ bw,
              float* __restrict__ pv, float* __restrict__ pb) {
  __shared__ float rv[256];
  __shared__ float rb[256];
  const int tid = threadIdx.x;
  const int i = blockIdx.x * 256 + tid;  // i < B_*L_ == 65536 exactly
  const int pi[6] = {0, 0, 0, 1, 1, 2};
  const int pj[6] = {1, 2, 3, 2, 3, 3};
  float xr[4];
#pragma unroll
  for (int c = 0; c < 4; ++c) xr[c] = x[(size_t)i * 4 + c];
  float v = 0.f, bsum = 0.f;
#pragma unroll
  for (int c = 0; c < 4; ++c) v += fabsf(xr[c] * vs[c] + vb[c]);
#pragma unroll
  for (int k = 0; k < 6; ++k) bsum += fabsf(bw[k] * (xr[pi[k]] - xr[pj[k]]));
  rv[tid] = v;
  rb[tid] = bsum;
  __syncthreads();
  for (int off = 128; off > 0; off >>= 1) {
    if (tid < off) { rv[tid] += rv[tid + off]; rb[tid] += rb[tid + off]; }
    __syncthreads();
  }
  if (tid == 0) { pv[blockIdx.x] = rv[0]; pb[blockIdx.x] = rb[0]; }
}

// ---------------------------------------------------------------------------
// fused embed + WMMA projection + WMMA banded-Gram attention for one tile
// variant 0 = base (writes predictions + weights); 1..3 = ablations (|diff|)
// ---------------------------------------------------------------------------
__global__ void __launch_bounds__(128)
ga_attn_kernel(const float* __restrict__ x, const float* __restrict__ sb,
               const float* __restrict__ vs, const float* __restrict__ vb,
               const float* __restrict__ bw, const float* __restrict__ WQ,
               const float* __restrict__ WK, const float* __restrict__ hb,
               const float* __restrict__ ws, float* __restrict__ predBase,
               float* __restrict__ diff, float* __restrict__ out, int variant) {
  __shared__ float Mlds[ROWS][16];
  __shared__ float Qlds[ROWS][H_];
  __shared__ float Klds[ROWS][H_];
  __shared__ float vh[ROWS];
  __shared__ float sc[TILE_T][LOOK];

  const int b = blockIdx.y;
  const int t0 = blockIdx.x * TILE_T;
  const int tid = threadIdx.x;
  const int zmask = (variant == 0) ? 0 : (1 << (variant - 1));

  // Warm caches for the weight rows this block's WMMA B-operands will gather
  // (lowers to global_prefetch_b8).
  __builtin_prefetch(WQ + (size_t)tid * DM_, 0, 0);
  __builtin_prefetch(WK + (size_t)tid * DM_, 0, 0);

  // Stage 1: embed 32 rows (rows t0 .. t0+31) into LDS, padded to 16 cols
  if (tid < ROWS) {
    const int grow = t0 + tid;
    float xr[4] = {0.f, 0.f, 0.f, 0.f};
    const bool ok = grow < L_;
    if (ok) {
#pragma unroll
      for (int c = 0; c < 4; ++c)
        xr[c] = x[((size_t)b * L_ + grow) * 4 + c];
    }
    float mr[16];
    embed_row(xr, sb[0], vs, vb, bw, ok ? zmask : 7, mr);
#pragma unroll
    for (int c = 0; c < 16; ++c) Mlds[tid][c] = mr[c];
  }
  __syncthreads();

  // Stage 2a: vh[row] = M_row . wv_head   (wave 0 only -> no intra-wave div)
  if (tid < ROWS) {
    float s = 0.f;
#pragma unroll
    for (int d = 0; d < DM_; ++d) s += Mlds[tid][d] * ws[WSO_WV + d];
    vh[tid] = s;
  }

  // Stage 2b: WMMA projections Q = phi(M WQ^T), K = phi(M WK^T)
  // 32 output tiles of 16x16 (2 matrices x 2 row-tiles x 8 col-tiles),
  // 8 tiles per wave; K-dim padded 11 -> 12 (3 steps of V_WMMA_F32_16X16X4_F32)
  const int wvid = tid >> 5;
  const int lane = tid & 31;
  for (int tile = wvid; tile < 32; tile += 4) {
    const int isK = tile >> 4;
    const int rt = (tile >> 3) & 1;
    const int ct = tile & 7;
    const float* W = isK ? WK : WQ;      // (H, Dm) row-major; B[k][n] = W[n][k]
    const int arow = rt * 16 + (lane & 15);
    const int ncol = ct * 16 + (lane & 15);
    v8f acc = {0.f, 0.f, 0.f, 0.f, 0.f, 0.f, 0.f, 0.f};
#pragma unroll
    for (int ks = 0; ks < 3; ++ks) {
      const int kk = ks * 4 + 2 * (lane >> 4);  // A 16x4 f32 layout
      v2f av, bv;
      av.x = Mlds[arow][kk];
      av.y = Mlds[arow][kk + 1];                // cols >= 11 are zero padding
      bv.x = (kk < DM_) ? W[ncol * DM_ + kk] : 0.f;
      bv.y = (kk + 1 < DM_) ? W[ncol * DM_ + kk + 1] : 0.f;
      acc = __builtin_amdgcn_wmma_f32_16x16x4_f32(false, av, false, bv,
                                                  (short)0, acc, false, false);
    }
    float* dst = isK ? &Klds[0][0] : &Qlds[0][0];
    const int rbase = rt * 16 + ((lane >> 4) << 3);  // D: vgpr r -> M=r(+8)
    const int col = ct * 16 + (lane & 15);
#pragma unroll
    for (int r = 0; r < 8; ++r)
      dst[(rbase + r) * H_ + col] = phi_f(acc[r]);
  }
  __syncthreads();

  // Stage 3: banded Gram via WMMA: G[i][j] = q(8+i) . k(j), keep l=j-i in [0,8)
  // Needed 16x16 tiles over (i,j) in [0,32)^2: (rt,ct) = (0,0), (0,1), (1,1).
  // Waves 0..2 take one tile each (uniform whole-wave branch, EXEC all ones).
  if (wvid < 3) {
    const int rt = (wvid == 2) ? 1 : 0;
    const int ct = (wvid == 0) ? 0 : 1;
    // A row = Q[8 + i]; rows 8+i > 31 are padding -> clamp (results unused)
    const int qrow0 = 8 + rt * 16 + (lane & 15);
    const int qrow = (qrow0 < ROWS) ? qrow0 : (ROWS - 1);
    const int krow = ct * 16 + (lane & 15);          // B[k][j] = K[j][k]
    v8f acc = {0.f, 0.f, 0.f, 0.f, 0.f, 0.f, 0.f, 0.f};
#pragma unroll
    for (int ks = 0; ks < H_ / 4; ++ks) {            // 32 k-steps over H=128
      const int kk = ks * 4 + 2 * (lane >> 4);
      v2f av, bv;
      av.x = Qlds[qrow][kk];
      av.y = Qlds[qrow][kk + 1];
      bv.x = Klds[krow][kk];
      bv.y = Klds[krow][kk + 1];
      acc = __builtin_amdgcn_wmma_f32_16x16x4_f32(false, av, false, bv,
                                                  (short)0, acc, false, false);
    }
    const int ibase = rt * 16 + ((lane >> 4) << 3);
    const int j = ct * 16 + (lane & 15);
#pragma unroll
    for (int r = 0; r < 8; ++r) {
      const int i = ibase + r;
      const int l = j - i;
      if (i < TILE_T && l >= 0 && l < LOOK) sc[i][l] = acc[r];
    }
  }
  __syncthreads();

  // Stage 4: S = sum(scores)+eps (== q.Z+eps), pred, weights / diffs
  if (tid < TILE_T) {
    const int t = t0 + tid;
    if (t < T_) {
      float S = EPSF, num = 0.f;
#pragma unroll
      for (int l = 0; l < LOOK; ++l) {
        S += sc[tid][l];
        num += sc[tid][l] * vh[tid + l];
      }
      const float pred = num / S + hb[0];
      const size_t oi = (size_t)b * T_ + t;
      if (variant == 0) {
        out[oi] = pred;
        predBase[oi] = pred;
#pragma unroll
        for (int l = 0; l < LOOK; ++l)
          out[(size_t)NPRED + oi * 8 + l] = sc[tid][l] / S;
      } else {
        diff[oi] = fabsf(predBase[oi] - pred);
      }
    }
  }
}

// ---------------------------------------------------------------------------
// qonly deltas: last window only (rows L-9..L-1), 3 masked-q variants
// ---------------------------------------------------------------------------
__global__ void __launch_bounds__(128)
ga_qonly_kernel(const float* __restrict__ x, const float* __restrict__ sb,
                const float* __restrict__ vs, const float* __restrict__ vb,
                const float* __restrict__ bw, const float* __restrict__ WQ,
                const float* __restrict__ WK, const float* __restrict__ hb,
                const float* __restrict__ ws, float* __restrict__ qpart) {
  __shared__ float M9[9][16];
  __shared__ float q[H_];
  __shared__ float kx[LOOK][H_];
  __shared__ float vh8[LOOK];
  __shared__ float tv[4][LOOK];
  const int b = blockIdx.x;
  const int tid = threadIdx.x;
  if (tid < 9) {
    const int row = L_ - 9 + tid;
    float xr[4];
#pragma unroll
    for (int c = 0; c < 4; ++c)
      xr[c] = x[((size_t)b * L_ + row) * 4 + c];
    float mr[16];
    embed_row(xr, sb[0], vs, vb, bw, 0, mr);
#pragma unroll
    for (int c = 0; c < 16; ++c) M9[tid][c] = mr[c];
  }
  __syncthreads();
  {
    float s = 0.f;
#pragma unroll
    for (int d = 0; d < DM_; ++d) s += M9[8][d] * WQ[tid * DM_ + d];
    q[tid] = phi_f(s);
    for (int l = 0; l < LOOK; ++l) {
      float t = 0.f;
#pragma unroll
      for (int d = 0; d < DM_; ++d) t += M9[l][d] * WK[tid * DM_ + d];
      kx[l][tid] = phi_f(t);
    }
  }
  if (tid < LOOK) {
    float s = 0.f;
#pragma unroll
    for (int d = 0; d < DM_; ++d) s += M9[tid][d] * ws[WSO_WV + d];
    vh8[tid] = s;
  }
  __syncthreads();
  if (tid < 32) {
    const int v = tid >> 3, l = tid & 7;
    float s = 0.f;
    for (int h = 0; h < H_; ++h) {
      float qh = q[h];
      if (v) qh *= (1.f - ws[WSO_HM + (v - 1) * H_ + h]);
      s += qh * kx[l][h];
    }
    tv[v][l] = s;
  }
  __syncthreads();
  if (tid == 0) {
    float o[4];
    for (int v = 0; v < 4; ++v) {
      float S = EPSF, num = 0.f;
#pragma unroll
      for (int l = 0; l < LOOK; ++l) { S += tv[v][l]; num += tv[v][l] * vh8[l]; }
      o[v] = num / S + hb[0];
    }
    for (int r = 0; r < 3; ++r) qpart[b * 3 + r] = fabsf(o[0] - o[r + 1]);
  }
}

// ---------------------------------------------------------------------------
// finalize: deterministic reductions -> 9 interpretability scalars
// ---------------------------------------------------------------------------
__global__ void __launch_bounds__(256)
ga_fin_kernel(const float* __restrict__ ws, const float* __restrict__ sb,
              float* __restrict__ out9) {
  __shared__ float red[256];
  const int tid = threadIdx.x;
  float dsum[3];
  for (int v = 0; v < 3; ++v) {
    const float* dif = ws + WSO_DIFF + (size_t)v * NPRED;
    float s = 0.f;
    for (int i = tid; i < NPRED; i += 256) s += dif[i];
    red[tid] = s;
    __syncthreads();
    for (int off = 128; off > 0; off >>= 1) {
      if (tid < off) red[tid] += red[tid + off];
      __syncthreads();
    }
    dsum[v] = red[0];
    __syncthreads();
  }
  if (tid == 0) {
    float vsum = 0.f, bsum = 0.f;
    for (int i = 0; i < 256; ++i) { vsum += ws[WSO_PV + i]; bsum += ws[WSO_PB + i]; }
    float qd[3] = {0.f, 0.f, 0.f};
    for (int b = 0; b < B_; ++b)
      for (int r = 0; r < 3; ++r) qd[r] += ws[WSO_QP + b * 3 + r];
    out9[0] = fabsf(1.f + sb[0]);
    out9[1] = vsum / (float)(B_ * L_ * 4);
    out9[2] = bsum / (float)(B_ * L_ * 6);
    for (int v = 0; v < 3; ++v) out9[3 + v] = dsum[v] / (float)NPRED;
    for (int r = 0; r < 3; ++r) out9[6 + r] = qd[r] / (float)B_;
  }
}

// ---------------------------------------------------------------------------
extern "C" void kernel_launch(void* const* d_in, const int* in_sizes, int n_in,
                              void* d_out, int out_size, void* d_ws,
                              size_t ws_size, hipStream_t stream) {
  (void)in_sizes; (void)n_in; (void)out_size; (void)ws_size;
  const float* x  = (const float*)d_in[0];
  const float* sb = (const float*)d_in[1];
  const float* vs = (const float*)d_in[2];
  const float* vb = (const float*)d_in[3];
  const float* bw = (const float*)d_in[4];
  const float* WQ = (const float*)d_in[5];
  const float* WK = (const float*)d_in[6];
  const float* WV = (const float*)d_in[7];
  const float* hW = (const float*)d_in[8];
  const float* hb = (const float*)d_in[9];
  float* ws = (float*)d_ws;
  float* out = (float*)d_out;

  float* pv = ws + WSO_PV;
  float* pb = ws + WSO_PB;
  float* qp = ws + WSO_QP;
  float* predBase = ws + WSO_PRED;

  ga_prep_kernel<<<1, 128, 0, stream>>>(WQ, WV, hW, ws);
  ga_mag_kernel<<<256, 256, 0, stream>>>(x, vs, vb, bw, pv, pb);
  for (int v = 0; v < 4; ++v) {
    float* diff = (v == 0) ? predBase : ws + WSO_DIFF + (size_t)(v - 1) * NPRED;
    ga_attn_kernel<<<dim3(NTILES, B_), 128, 0, stream>>>(
        x, sb, vs, vb, bw, WQ, WK, hb, ws, predBase, diff, out, v);
  }
  ga_qonly_kernel<<<B_, 128, 0, stream>>>(x, sb, vs, vb, bw, WQ, WK, hb, ws, qp);
  ga_fin_kernel<<<1, 256, 0, stream>>>(ws, sb, out + (size_t)NPRED * 9);
}